// GCNNet_13262859010221
// MI455X (gfx1250) — compile-verified
//
#include <hip/hip_runtime.h>
#include <math.h>

typedef __attribute__((ext_vector_type(2))) float v2f;
typedef __attribute__((ext_vector_type(8))) float v8f;

#define N_NODES 50000
#define N_EDGES 800000
#define F_IN    256
#define F_HID   128
#define F_OUT   40
#define F_PAD   48   // N padded to 3 WMMA tiles for layer 2

// ---------------- degree / normalization ----------------
__global__ void k_init_deg(float* __restrict__ deg) {
  int i = blockIdx.x * blockDim.x + threadIdx.x;
  if (i < N_NODES) deg[i] = 1.0f;                       // self-loop
}

__global__ void k_count_deg(const int* __restrict__ ei, float* __restrict__ deg) {
  int e = blockIdx.x * blockDim.x + threadIdx.x;
  if (e < N_EDGES) atomicAdd(deg + ei[N_EDGES + e], 1.0f);   // col = ei[1][e]
}

__global__ void k_dinv(const float* __restrict__ deg, float* __restrict__ dinv) {
  int i = blockIdx.x * blockDim.x + threadIdx.x;
  if (i < N_NODES) dinv[i] = rsqrtf(deg[i]);            // deg >= 1 always
}

// ---------------- layer 1 GEMM: hs = dinv .* (x @ W1), agg1 init = hs ----------
// Wave computes a 16x128 slab: 8 N-tiles of V_WMMA_F32_16X16X4_F32, K looped 4-at-a-time.
// W1 staged transposed in LDS in 64-wide K chunks; stride 68 => conflict-free b64 reads.
__global__ __launch_bounds__(256) void k_gemm1(const float* __restrict__ x,
                                               const float* __restrict__ W1,
                                               const float* __restrict__ dinv,
                                               float* __restrict__ hs,
                                               float* __restrict__ agg1) {
  __shared__ float w1t[F_HID][68];                       // [col][k_local], 34 KB
  const int lane = threadIdx.x & 31;
  const int wave = threadIdx.x >> 5;
  const int half = lane >> 4;                            // 0: K even pair, 1: K odd pair
  const int l16  = lane & 15;
  const int row0 = (blockIdx.x * 8 + wave) * 16;
  const int rowA = min(row0 + l16, N_NODES - 1);         // clamp tail reads

  const v8f zero8 = {0.f,0.f,0.f,0.f,0.f,0.f,0.f,0.f};
  v8f acc[8];
  #pragma unroll
  for (int j = 0; j < 8; ++j) acc[j] = zero8;

  for (int kc = 0; kc < F_IN; kc += 64) {
    __syncthreads();
    for (int g = threadIdx.x; g < 64 * F_HID; g += 256) {
      int kl = g >> 7;                                   // coalesced over c
      int c  = g & 127;
      w1t[c][kl] = W1[(size_t)(kc + kl) * F_HID + c];
    }
    __syncthreads();
    #pragma unroll
    for (int k4 = 0; k4 < 16; ++k4) {
      const int klocal = k4 * 4 + half * 2;
      // A frag (ISA 16x4 f32 layout): lane l16 row, this half-wave's K pair
      v2f a = *(const v2f*)(x + (size_t)rowA * F_IN + kc + klocal);
      #pragma unroll
      for (int j = 0; j < 8; ++j) {
        v2f b = *(const v2f*)(&w1t[j * 16 + l16][klocal]);
        acc[j] = __builtin_amdgcn_wmma_f32_16x16x4_f32(
            false, a, false, b, (short)0, acc[j], false, false);
      }
    }
  }
  // epilogue: C/D layout => VGPR r holds (M = r + 8*half, N = l16)
  #pragma unroll
  for (int r = 0; r < 8; ++r) {
    int row = row0 + r + half * 8;
    if (row < N_NODES) {
      float d = dinv[row];
      #pragma unroll
      for (int j = 0; j < 8; ++j) {
        float v = d * acc[j][r];
        size_t o = (size_t)row * F_HID + j * 16 + l16;
        hs[o]   = v;                                      // gather source
        agg1[o] = v;                                      // self-loop seed
      }
    }
  }
}

// ---------------- layer 1 scatter: agg1[col] += hs[row] -------------------------
__global__ void k_scatter1(const int* __restrict__ ei, const float* __restrict__ hs,
                           float* __restrict__ agg1) {
  unsigned idx = blockIdx.x * blockDim.x + threadIdx.x;  // E*128 = 102.4M
  int e = idx >> 7, f = idx & 127;
  int r = ei[e];
  int c = ei[N_EDGES + e];
  atomicAdd(agg1 + (size_t)c * F_HID + f, hs[(size_t)r * F_HID + f]);
}

// ---------------- layer 1 combine: a1 = relu(dinv * agg1 + b1) ------------------
__global__ void k_combine1(const float* __restrict__ agg1, const float* __restrict__ dinv,
                           const float* __restrict__ b1, float* __restrict__ a1) {
  int idx = blockIdx.x * blockDim.x + threadIdx.x;
  if (idx < N_NODES * F_HID) {
    int i = idx >> 7, f = idx & 127;
    float v = dinv[i] * agg1[idx] + b1[f];
    a1[idx] = v > 0.f ? v : 0.f;
  }
}

// ---------------- layer 2 GEMM: gs = dinv .* (a1 @ W2), agg2 init = gs ----------
__global__ __launch_bounds__(256) void k_gemm2(const float* __restrict__ a1,
                                               const float* __restrict__ W2,
                                               const float* __restrict__ dinv,
                                               float* __restrict__ gs,
                                               float* __restrict__ agg2) {
  __shared__ float w2t[F_PAD][132];                      // [col][k], 25 KB, pad cols 40..47
  const int lane = threadIdx.x & 31;
  const int wave = threadIdx.x >> 5;
  const int half = lane >> 4;
  const int l16  = lane & 15;
  const int row0 = (blockIdx.x * 8 + wave) * 16;
  const int rowA = min(row0 + l16, N_NODES - 1);

  for (int g = threadIdx.x; g < F_PAD * F_HID; g += 256) {
    int c = g >> 7, k = g & 127;
    w2t[c][k] = (c < F_OUT) ? W2[(size_t)k * F_OUT + c] : 0.f;
  }
  __syncthreads();

  const v8f zero8 = {0.f,0.f,0.f,0.f,0.f,0.f,0.f,0.f};
  v8f acc[3];
  #pragma unroll
  for (int j = 0; j < 3; ++j) acc[j] = zero8;

  #pragma unroll
  for (int k4 = 0; k4 < 32; ++k4) {
    const int kk = k4 * 4 + half * 2;
    v2f a = *(const v2f*)(a1 + (size_t)rowA * F_HID + kk);
    #pragma unroll
    for (int j = 0; j < 3; ++j) {
      v2f b = *(const v2f*)(&w2t[j * 16 + l16][kk]);
      acc[j] = __builtin_amdgcn_wmma_f32_16x16x4_f32(
          false, a, false, b, (short)0, acc[j], false, false);
    }
  }
  #pragma unroll
  for (int r = 0; r < 8; ++r) {
    int row = row0 + r + half * 8;
    if (row < N_NODES) {
      float d = dinv[row];
      #pragma unroll
      for (int j = 0; j < 3; ++j) {
        int col = j * 16 + l16;
        if (col < F_OUT) {
          float v = d * acc[j][r];
          size_t o = (size_t)row * F_OUT + col;
          gs[o]   = v;
          agg2[o] = v;
        }
      }
    }
  }
}

// ---------------- layer 2 scatter -----------------------------------------------
__global__ void k_scatter2(const int* __restrict__ ei, const float* __restrict__ gs,
                           float* __restrict__ agg2) {
  int idx = blockIdx.x * blockDim.x + threadIdx.x;       // E*40 = 32M
  if (idx < N_EDGES * F_OUT) {
    int e = idx / F_OUT, f = idx - e * F_OUT;
    int r = ei[e];
    int c = ei[N_EDGES + e];
    atomicAdd(agg2 + (size_t)c * F_OUT + f, gs[(size_t)r * F_OUT + f]);
  }
}

// ---------------- layer 2 combine: logits (second output) -----------------------
__global__ void k_combine2(const float* __restrict__ agg2, const float* __restrict__ dinv,
                           const float* __restrict__ b2, float* __restrict__ logits) {
  int idx = blockIdx.x * blockDim.x + threadIdx.x;
  if (idx < N_NODES * F_OUT) {
    int i = idx / F_OUT, f = idx - i * F_OUT;
    logits[idx] = dinv[i] * agg2[idx] + b2[f];
  }
}

// ---------------- log_softmax over 40 classes: one wave32 per node --------------
__global__ __launch_bounds__(256) void k_logsoftmax(const float* __restrict__ logits,
                                                    float* __restrict__ out) {
  int lane = threadIdx.x & 31;
  int node = blockIdx.x * 8 + (threadIdx.x >> 5);
  if (node >= N_NODES) return;
  const float* L = logits + (size_t)node * F_OUT;
  float v0 = L[lane];                                    // lane < 32 < 40
  float v1 = (lane < F_OUT - 32) ? L[lane + 32] : -3.4e38f;
  float m = fmaxf(v0, v1);
  #pragma unroll
  for (int off = 16; off; off >>= 1) m = fmaxf(m, __shfl_xor(m, off, 32));
  float s = __expf(v0 - m) + ((lane < F_OUT - 32) ? __expf(v1 - m) : 0.f);
  #pragma unroll
  for (int off = 16; off; off >>= 1) s += __shfl_xor(s, off, 32);
  float lse = m + __logf(s);
  out[(size_t)node * F_OUT + lane] = v0 - lse;
  if (lane < F_OUT - 32) out[(size_t)node * F_OUT + lane + 32] = v1 - lse;
}

// ---------------- launch ---------------------------------------------------------
extern "C" void kernel_launch(void* const* d_in, const int* in_sizes, int n_in,
                              void* d_out, int out_size, void* d_ws, size_t ws_size,
                              hipStream_t stream) {
  const float* x  = (const float*)d_in[0];
  const int*   ei = (const int*)d_in[1];                 // [2, E] flat
  const float* W1 = (const float*)d_in[2];
  const float* b1 = (const float*)d_in[3];
  const float* W2 = (const float*)d_in[4];
  const float* b2 = (const float*)d_in[5];

  float* w    = (float*)d_ws;
  float* deg  = w;                                       //  50,000
  float* dinv = w + 50000;                               //  50,000
  float* hs   = w + 100000;                              // 6,400,000 (later reused as a1)
  float* agg1 = w + 6500000;                             // 6,400,000 (later reused as gs)
  float* agg2 = w + 12900000;                            // 2,000,000
  float* a1   = hs;                                      // combine1 overwrites hs region
  float* gs   = agg1;                                    // gemm2 overwrites agg1 region

  float* log_probs = (float*)d_out;                      // output 0: [50000,40]
  float* logits    = (float*)d_out + (size_t)N_NODES * F_OUT;  // output 1: [50000,40]

  const int T = 256;
  k_init_deg  <<<(N_NODES + T - 1) / T, T, 0, stream>>>(deg);
  k_count_deg <<<(N_EDGES + T - 1) / T, T, 0, stream>>>(ei, deg);
  k_dinv      <<<(N_NODES + T - 1) / T, T, 0, stream>>>(deg, dinv);

  const int gemm_blocks = (N_NODES + 127) / 128;         // 8 waves x 16 rows per block
  k_gemm1<<<gemm_blocks, T, 0, stream>>>(x, W1, dinv, hs, agg1);
  k_scatter1<<<(unsigned)((size_t)N_EDGES * F_HID / T), T, 0, stream>>>(ei, hs, agg1);
  k_combine1<<<(N_NODES * F_HID + T - 1) / T, T, 0, stream>>>(agg1, dinv, b1, a1);

  k_gemm2<<<gemm_blocks, T, 0, stream>>>(a1, W2, dinv, gs, agg2);
  k_scatter2<<<(N_EDGES * F_OUT + T - 1) / T, T, 0, stream>>>(ei, gs, agg2);
  k_combine2<<<(N_NODES * F_OUT + T - 1) / T, T, 0, stream>>>(agg2, dinv, b2, logits);

  k_logsoftmax<<<(N_NODES + 7) / 8, T, 0, stream>>>(logits, log_probs);
}